// GAT_43164421325046
// MI455X (gfx1250) — compile-verified
//
#include <hip/hip_runtime.h>
#include <hip/hip_bf16.h>
#include <math.h>

typedef __attribute__((ext_vector_type(2))) float v2f;
typedef __attribute__((ext_vector_type(4))) float v4f;
typedef __attribute__((ext_vector_type(8))) float v8f;
typedef int gv4i __attribute__((vector_size(16)));   // matches builtin's pointee type

#define F_IN  256
#define F_OUT 32
#define NEG_SLOPE 0.2f
#define XS 68    // padded LDS row stride (floats) for x tile chunk (272B, 16B aligned)
#define WSTR 258 // padded LDS row stride (floats) for transposed W (1032B, 8B aligned)

#define AS1 __attribute__((address_space(1)))
#define AS3 __attribute__((address_space(3)))

// CDNA5 async global->LDS copy (ASYNCcnt-tracked), with graceful fallback.
#if defined(__AMDGCN__) && __has_builtin(__builtin_amdgcn_global_load_async_to_lds_b128)
#define HAVE_ASYNC_LDS 1
#else
#define HAVE_ASYNC_LDS 0
#endif

__device__ __forceinline__ void wait_async_lds() {
#if defined(__AMDGCN__)
#if __has_builtin(__builtin_amdgcn_s_wait_asynccnt)
  __builtin_amdgcn_s_wait_asynccnt(0);
#else
  asm volatile("s_wait_asynccnt 0x0" ::: "memory");
#endif
#endif
}

// monotonic float <-> orderable-uint encoding for atomic max on floats
__device__ __forceinline__ unsigned fenc(float f) {
  unsigned u = __float_as_uint(f);
  return (u & 0x80000000u) ? ~u : (u | 0x80000000u);
}
__device__ __forceinline__ float fdec(unsigned u) {
  unsigned v = (u & 0x80000000u) ? (u & 0x7FFFFFFFu) : ~u;
  return __uint_as_float(v);
}

// -------- Kernel 1: h = x @ W via V_WMMA_F32_16X16X4_F32 --------
// 256 threads = 8 wave32. Block covers 64 rows; wave w -> rowTile = w>>1,
// colTile = w&1 (cols 0-15 or 16-31). K staged in chunks of 64, moved
// global->LDS with GLOBAL_LOAD_ASYNC_TO_LDS_B128 when available.
__global__ __launch_bounds__(256) void gat_gemm_wmma(
    const float* __restrict__ x, const float* __restrict__ W,
    float* __restrict__ h, int n)
{
  __shared__ float sWt[F_OUT * WSTR];   // W transposed: [col][k]
  __shared__ float sX[4 * 16 * XS];     // 4 row-tiles x 16 rows x 64 k-chunk

  const int tid  = threadIdx.x;
  const int wave = tid >> 5;
  const int lane = tid & 31;
  const int l16  = lane & 15;
  const int half = lane >> 4;           // 0: K{0,1}, 1: K{2,3} per ISA A/B layout
  const int rowTile = wave >> 1;
  const int colTile = wave & 1;
  const int rowBase = blockIdx.x * 64;

  // Stage W transposed (coalesced read of W[k*32+c])
  for (int i = tid; i < F_IN * F_OUT; i += 256) {
    int k = i >> 5;
    int c = i & 31;
    sWt[c * WSTR + k] = W[i];
  }

  v8f acc = {0.f, 0.f, 0.f, 0.f, 0.f, 0.f, 0.f, 0.f};
  const float* swc = &sWt[(colTile * 16 + l16) * WSTR];

  for (int k0 = 0; k0 < F_IN; k0 += 64) {
    __syncthreads();   // previous chunk's readers done / W staging visible
    // Stage x chunk: 4 tiles x 16 rows x 64 cols = 1024 b128 transfers
    #pragma unroll
    for (int it = 0; it < 4; ++it) {
      int f    = tid + it * 256;        // 0..1023
      int tile = f >> 8;
      int row  = (f >> 4) & 15;
      int c4   = f & 15;
      int gr   = rowBase + tile * 16 + row;
      if (gr >= n) gr = n - 1;          // clamp tail rows (stores guarded later)
      const float* gp = &x[(size_t)gr * F_IN + k0 + c4 * 4];
      float*       lp = &sX[(tile * 16 + row) * XS + c4 * 4];
#if HAVE_ASYNC_LDS
      __builtin_amdgcn_global_load_async_to_lds_b128(
          (AS1 gv4i*)gp, (AS3 gv4i*)lp, 0, 0);
#else
      *(v4f*)lp = *(const v4f*)gp;
#endif
    }
#if HAVE_ASYNC_LDS
    wait_async_lds();                   // ASYNCcnt==0 -> LDS writes landed
#endif
    __syncthreads();

    // Prefetch next chunk of this wave's rows into cache (global_prefetch_b8)
    if (k0 + 64 < F_IN) {
      int gr = rowBase + rowTile * 16 + l16;
      if (gr >= n) gr = n - 1;
      __builtin_prefetch(&x[(size_t)gr * F_IN + k0 + 64 + half * 32], 0, 1);
    }

    const float* sxr = &sX[rowTile * 16 * XS];
    #pragma unroll
    for (int kk = 0; kk < 64; kk += 4) {
      // A: 16x4 f32 frag — lane l16 holds row l16, K = kk+2*half .. +1
      v2f a = *(const v2f*)&sxr[l16 * XS + kk + 2 * half];
      // B: 4x16 f32 frag — lane l16 holds col, same K striping
      v2f b = *(const v2f*)&swc[k0 + kk + 2 * half];
      acc = __builtin_amdgcn_wmma_f32_16x16x4_f32(
          false, a, false, b, (short)0, acc, false, false);
    }
  }

  // D layout: VGPR r -> rows M=r (lanes 0-15) / M=r+8 (lanes 16-31), col = l16
  const int col = colTile * 16 + l16;
  #pragma unroll
  for (int r = 0; r < 8; ++r) {
    int row = rowBase + rowTile * 16 + r + 8 * half;
    if (row < n) h[(size_t)row * F_OUT + col] = acc[r];
  }
}

// -------- Kernel 2: per-node attention logits --------
__global__ void gat_att(const float* __restrict__ h,
                        const float* __restrict__ att_src,
                        const float* __restrict__ att_dst,
                        float* __restrict__ a_s, float* __restrict__ a_d, int n)
{
  int i = blockIdx.x * blockDim.x + threadIdx.x;
  if (i >= n) return;
  const float* hr = &h[(size_t)i * F_OUT];
  float s = 0.f, d = 0.f;
  #pragma unroll
  for (int c = 0; c < F_OUT; ++c) {
    float v = hr[c];
    s += v * att_src[c];
    d += v * att_dst[c];
  }
  a_s[i] = s;
  a_d[i] = d;
}

// -------- Kernel 3: init accumulators --------
__global__ void gat_init(float* __restrict__ out, unsigned* __restrict__ menc,
                         float* __restrict__ denom, int n)
{
  int i = blockIdx.x * blockDim.x + threadIdx.x;
  if (i < n * F_OUT) out[i] = 0.f;
  if (i < n) { menc[i] = 0u; denom[i] = 0.f; }
}

// -------- Kernel 4: edge logits + segment max (atomic on encoded uint) ----
__global__ void gat_logit_max(const int* __restrict__ ei,
                              const float* __restrict__ a_s,
                              const float* __restrict__ a_d,
                              float* __restrict__ ew, unsigned* __restrict__ menc,
                              int n, int e)
{
  int idx = blockIdx.x * blockDim.x + threadIdx.x;
  int tot = e + n;
  if (idx >= tot) return;
  int s, d;
  if (idx < e) { s = ei[idx]; d = ei[e + idx]; }
  else         { s = d = idx - e; }            // self-loops appended
  float z = a_s[s] + a_d[d];
  z = (z > 0.f) ? z : NEG_SLOPE * z;           // leaky relu
  ew[idx] = z;
  atomicMax(&menc[d], fenc(z));
}

// -------- Kernel 5: exp(logit - max) + segment sum --------
__global__ void gat_exp_sum(const int* __restrict__ ei,
                            float* __restrict__ ew,
                            const unsigned* __restrict__ menc,
                            float* __restrict__ denom, int n, int e)
{
  int idx = blockIdx.x * blockDim.x + threadIdx.x;
  int tot = e + n;
  if (idx >= tot) return;
  int d = (idx < e) ? ei[e + idx] : (idx - e);
  float w = __expf(ew[idx] - fdec(menc[d]));
  ew[idx] = w;
  atomicAdd(&denom[d], w);
}

// -------- Kernel 6: weighted scatter-add aggregation --------
__global__ void gat_aggregate(const int* __restrict__ ei,
                              const float* __restrict__ h,
                              const float* __restrict__ ew,
                              const float* __restrict__ denom,
                              float* __restrict__ out, int n, int e)
{
  int idx = blockIdx.x * blockDim.x + threadIdx.x;
  int tot = e + n;
  if (idx >= tot) return;
  int s, d;
  if (idx < e) { s = ei[idx]; d = ei[e + idx]; }
  else         { s = d = idx - e; }
  float coef = ew[idx] / denom[d];
  const float* hs = &h[(size_t)s * F_OUT];
  float* od = &out[(size_t)d * F_OUT];
  #pragma unroll
  for (int c = 0; c < F_OUT; ++c)
    atomicAdd(&od[c], hs[c] * coef);
}

// -------- Kernel 7: bias + ReLU + dropout(p=0.5, deterministic hash) ------
__global__ void gat_finalize(float* __restrict__ out,
                             const float* __restrict__ bias, int total)
{
  int idx = blockIdx.x * blockDim.x + threadIdx.x;
  if (idx >= total) return;
  float v = out[idx] + bias[idx & (F_OUT - 1)];
  v = (v > 0.f) ? v : 0.f;
  unsigned hsh = (unsigned)idx * 2654435761u + 0x9E3779B9u;
  hsh ^= hsh >> 16; hsh *= 0x85EBCA6Bu; hsh ^= hsh >> 13;
  out[idx] = (hsh & 1u) ? v * 2.0f : 0.0f;   // keep-prob 0.5, scale 1/(1-p)
}

extern "C" void kernel_launch(void* const* d_in, const int* in_sizes, int n_in,
                              void* d_out, int out_size, void* d_ws, size_t ws_size,
                              hipStream_t stream)
{
  const float* x       = (const float*)d_in[0];
  const float* W       = (const float*)d_in[1];
  const float* att_src = (const float*)d_in[2];
  const float* att_dst = (const float*)d_in[3];
  const float* bias    = (const float*)d_in[4];
  const int*   ei      = (const int*)d_in[5];

  const int n = in_sizes[0] / F_IN;     // 100000
  const int e = in_sizes[5] / 2;        // 1600000
  const int tot = e + n;
  float* out = (float*)d_out;

  // Workspace layout
  float*    h     = (float*)d_ws;                     // n*32
  float*    a_s   = h + (size_t)n * F_OUT;            // n
  float*    a_d   = a_s + n;                          // n
  unsigned* menc  = (unsigned*)(a_d + n);             // n
  float*    denom = (float*)(menc + n);               // n
  float*    ew    = denom + n;                        // e+n

  dim3 blk(256);
  gat_gemm_wmma<<<dim3((n + 63) / 64), blk, 0, stream>>>(x, W, h, n);
  gat_att<<<dim3((n + 255) / 256), blk, 0, stream>>>(h, att_src, att_dst, a_s, a_d, n);
  gat_init<<<dim3((n * F_OUT + 255) / 256), blk, 0, stream>>>(out, menc, denom, n);
  gat_logit_max<<<dim3((tot + 255) / 256), blk, 0, stream>>>(ei, a_s, a_d, ew, menc, n, e);
  gat_exp_sum<<<dim3((tot + 255) / 256), blk, 0, stream>>>(ei, ew, menc, denom, n, e);
  gat_aggregate<<<dim3((tot + 255) / 256), blk, 0, stream>>>(ei, h, ew, denom, out, n, e);
  gat_finalize<<<dim3((n * F_OUT + 255) / 256), blk, 0, stream>>>(out, bias, n * F_OUT);
}